// GLSTMCell_8435315769497
// MI455X (gfx1250) — compile-verified
//
#include <hip/hip_runtime.h>
#include <cmath>
#include <cstdint>

// GLSTM cell for MI455X (gfx1250). HBM-bound (~272MB moved, 34.4 GFLOP):
// f32 WMMA (V_WMMA_F32_16X16X4_F32) keeps full precision while exceeding the
// ~2.3 TFLOPS roofline balance point by a wide margin. h is read from HBM
// exactly once (mean over W fused into GEMM staging). Tile staging uses
// GLOBAL_LOAD_ASYNC_TO_LDS_B128 + ASYNCcnt double-buffering when available.

typedef float v2f __attribute__((ext_vector_type(2)));
typedef float v8f __attribute__((ext_vector_type(8)));

constexpr int B  = 128;
constexpr int W  = 512;
constexpr int H  = 512;
constexpr int H2 = 2 * H;
constexpr int LSTR = 36;  // LDS row stride (floats): mult of 4 -> b128 stores,
                          // even -> 8B-aligned b64 fragment reads.

#if __has_builtin(__builtin_amdgcn_global_load_async_to_lds_b128)
#define ASYNC_STAGE 1
#else
#define ASYNC_STAGE 0
#endif

#if ASYNC_STAGE
// The builtin takes pointers to 16-byte int vectors: global src, LDS dst.
typedef int b128v __attribute__((vector_size(16)));
typedef __attribute__((address_space(1))) b128v* as1_b128p;
typedef __attribute__((address_space(3))) b128v* as3_b128p;

__device__ __forceinline__ void async_b128(const float* gsrc, float* lds) {
    // generic shared ptr: low 32 bits are the LDS byte offset
    as1_b128p g = (as1_b128p)(unsigned long long)(uintptr_t)gsrc;
    as3_b128p l = (as3_b128p)(unsigned int)(uintptr_t)lds;
    __builtin_amdgcn_global_load_async_to_lds_b128(g, l, 0, 0);
}
__device__ __forceinline__ void wait_async_le4() {
#if __has_builtin(__builtin_amdgcn_s_wait_asynccnt)
    __builtin_amdgcn_s_wait_asynccnt(4);
#else
    asm volatile("s_wait_asynccnt 0x4" ::: "memory");
#endif
}
__device__ __forceinline__ void wait_async_0() {
#if __has_builtin(__builtin_amdgcn_s_wait_asynccnt)
    __builtin_amdgcn_s_wait_asynccnt(0);
#else
    asm volatile("s_wait_asynccnt 0x0" ::: "memory");
#endif
}
#endif

// ---------------------------------------------------------------------------
// Kernel 1: gw[b][n] = g[b]·w_w[n]  (no h_avg dependency, so it runs first)
// and zero the three atomic accumulator arrays (d_ws is poisoned 0xAA).
// ---------------------------------------------------------------------------
__global__ void k_gw_zero(const float* __restrict__ g,
                          const float* __restrict__ w_w,
                          float* __restrict__ gw,
                          float* __restrict__ h_avg_sum,
                          float* __restrict__ sum_e,
                          float* __restrict__ sum_ce) {
    int tid = blockIdx.x * blockDim.x + threadIdx.x;
    if (tid >= B * H) return;
    int b = tid / H, n = tid % H;
    const float* grow = g + (size_t)b * H;
    const float* wr   = w_w + (size_t)n * H;
    float acc = 0.f;
#pragma unroll 4
    for (int k = 0; k < H; ++k) acc += grow[k] * wr[k];
    gw[tid]        = acc;
    h_avg_sum[tid] = 0.f;
    sum_e[tid]     = 0.f;
    sum_ce[tid]    = 0.f;
}

// ---------------------------------------------------------------------------
// Kernel 2: the big fused GEMM (one HBM pass over h and c).
//   t[w,n]  = sum_k h[b,w,k] * u_w[n,k]              (WMMA f32 16x16x4)
//   e       = exp(sigmoid(t + gw[b,n] + u_b[n]))     (sigmoid in (0,1) =>
//                                                     exp safe, no max pass)
//   sum_e [b,n] += column-sum_w e
//   sum_ce[b,n] += column-sum_w c[b,w,n]*e
//   h_avg_sum[b,k] += column-sum_w h[b,w,k]          (n0==0 blocks only; they
//                                                     stage each h tile once)
// Workgroup = 256 threads (8 waves) -> 64(w) x 64(n) tile of batch b.
// Wave (wr, wg): rows wr*16..+15, two n-tiles wg*32 and wg*32+16 (A reused).
// ---------------------------------------------------------------------------
__global__ __launch_bounds__(256) void k_fw_wmma(
        const float* __restrict__ h,
        const float* __restrict__ c_in,
        const float* __restrict__ u_w,
        const float* __restrict__ u_b,
        const float* __restrict__ gw,
        float* __restrict__ h_avg_sum,
        float* __restrict__ sum_e,
        float* __restrict__ sum_ce) {
    const int n0 = blockIdx.x * 64;
    const int w0 = blockIdx.y * 64;
    const int b  = blockIdx.z;

    const int t    = threadIdx.x;
    const int lane = t & 31;
    const int wave = t >> 5;
    const int id   = lane & 15;   // M (A) / N (B,C) within a 16x16 tile
    const int hs   = lane >> 4;   // half-select: K pair / M+8 select
    const int wr   = wave & 3;    // which 16-row band of w
    const int wg   = wave >> 2;   // 0/1: which pair of n-tiles

    const float* hbase = h + (size_t)b * W * H;

    // Vectorized staging indices: 64x32 tile = 512 float4s, 2 per thread.
    const int r_a = t >> 3,         c_a = (t & 7) * 4;         // elems   0..255
    const int r_b = (t + 256) >> 3, c_b = ((t + 256) & 7) * 4; // elems 256..511

    v8f acc0 = {};
    v8f acc1 = {};

#if ASYNC_STAGE
    // -------- async DMA staging with LDS double-buffering (ASYNCcnt) -------
    __shared__ __align__(16) float Ah[2][64][LSTR];
    __shared__ __align__(16) float Bu[2][64][LSTR];

    auto issue = [&](int k0, int buf) {
        async_b128(&hbase[(size_t)(w0 + r_a) * H + k0 + c_a], &Ah[buf][r_a][c_a]);
        async_b128(&hbase[(size_t)(w0 + r_b) * H + k0 + c_b], &Ah[buf][r_b][c_b]);
        async_b128(&u_w[(size_t)(n0 + r_a) * H + k0 + c_a],   &Bu[buf][r_a][c_a]);
        async_b128(&u_w[(size_t)(n0 + r_b) * H + k0 + c_b],   &Bu[buf][r_b][c_b]);
    };

    issue(0, 0);
    for (int i = 0; i < H / 32; ++i) {
        const int k0  = i * 32;
        const int cur = i & 1;
        __syncthreads();  // all waves finished reading buf[cur^1] (from iter i-1)
        if (i + 1 < H / 32) {
            issue(k0 + 32, cur ^ 1);  // DMA next chunk during this compute
            wait_async_le4();         // in-order: chunk i's 4 loads complete
        } else {
            wait_async_0();
        }
        __syncthreads();  // chunk i visible to all waves

        if (n0 == 0) {  // fused h-mean (this block uniquely stages this tile)
            int col = t & 31;
            int r0  = (t >> 5) * 8;
            float s = 0.f;
#pragma unroll
            for (int j = 0; j < 8; ++j) s += Ah[cur][r0 + j][col];
            atomicAdd(&h_avg_sum[(size_t)b * H + k0 + col], s);
        }

#pragma unroll
        for (int kk = 0; kk < 32; kk += 4) {
            v2f a  = *(const v2f*)&Ah[cur][wr * 16 + id][kk + 2 * hs];
            v2f b0 = *(const v2f*)&Bu[cur][wg * 32 + id][kk + 2 * hs];
            v2f b1 = *(const v2f*)&Bu[cur][wg * 32 + 16 + id][kk + 2 * hs];
            acc0 = __builtin_amdgcn_wmma_f32_16x16x4_f32(
                false, a, false, b0, (short)0, acc0, false, false);
            acc1 = __builtin_amdgcn_wmma_f32_16x16x4_f32(
                false, a, false, b1, (short)0, acc1, false, false);
        }
    }
#else
    // -------- fallback: synchronous b128 staging (proven path) -------------
    __shared__ __align__(16) float Ah[64][LSTR];
    __shared__ __align__(16) float Bu[64][LSTR];

    for (int k0 = 0; k0 < H; k0 += 32) {
        __syncthreads();
        {
            const float4* ga0 = (const float4*)&hbase[(size_t)(w0 + r_a) * H + k0 + c_a];
            const float4* ga1 = (const float4*)&hbase[(size_t)(w0 + r_b) * H + k0 + c_b];
            const float4* gb0 = (const float4*)&u_w[(size_t)(n0 + r_a) * H + k0 + c_a];
            const float4* gb1 = (const float4*)&u_w[(size_t)(n0 + r_b) * H + k0 + c_b];
            *(float4*)&Ah[r_a][c_a] = *ga0;
            *(float4*)&Ah[r_b][c_b] = *ga1;
            *(float4*)&Bu[r_a][c_a] = *gb0;
            *(float4*)&Bu[r_b][c_b] = *gb1;
            if (k0 + 32 < H) {
                __builtin_prefetch(ga0 + 8, 0, 3);
                __builtin_prefetch(ga1 + 8, 0, 3);
                __builtin_prefetch(gb0 + 8, 0, 3);
                __builtin_prefetch(gb1 + 8, 0, 3);
            }
        }
        __syncthreads();

        if (n0 == 0) {
            int col = t & 31;
            int r0  = (t >> 5) * 8;
            float s = 0.f;
#pragma unroll
            for (int j = 0; j < 8; ++j) s += Ah[r0 + j][col];
            atomicAdd(&h_avg_sum[(size_t)b * H + k0 + col], s);
        }

#pragma unroll
        for (int kk = 0; kk < 32; kk += 4) {
            v2f a  = *(const v2f*)&Ah[wr * 16 + id][kk + 2 * hs];
            v2f b0 = *(const v2f*)&Bu[wg * 32 + id][kk + 2 * hs];
            v2f b1 = *(const v2f*)&Bu[wg * 32 + 16 + id][kk + 2 * hs];
            acc0 = __builtin_amdgcn_wmma_f32_16x16x4_f32(
                false, a, false, b0, (short)0, acc0, false, false);
            acc1 = __builtin_amdgcn_wmma_f32_16x16x4_f32(
                false, a, false, b1, (short)0, acc1, false, false);
        }
    }
#endif

    // Fused epilogue: sigmoid -> exp -> c-weighted column reduction.
    // C/D layout: VGPR j holds (M = j + 8*hs, N = id).
#pragma unroll
    for (int tt = 0; tt < 2; ++tt) {
        v8f acc = tt ? acc1 : acc0;
        const int n_glob = n0 + wg * 32 + tt * 16 + id;
        const float bias = gw[(size_t)b * H + n_glob] + u_b[n_glob];
        float pe = 0.f, pce = 0.f;
#pragma unroll
        for (int j = 0; j < 8; ++j) {
            int wrow  = w0 + wr * 16 + hs * 8 + j;
            float pre = acc[j] + bias;
            float s   = 1.0f / (1.0f + __expf(-pre));
            float e   = __expf(s);  // s in (0,1): always safe without max pass
            float cin = c_in[(size_t)b * W * H + (size_t)wrow * H + n_glob];
            pe  += e;
            pce += cin * e;
        }
        // fold the two 8-row halves (lane l <-> lane l+16 share column N=id)
        pe  += __shfl_xor(pe, 16, 32);
        pce += __shfl_xor(pce, 16, 32);
        if (hs == 0) {
            atomicAdd(&sum_e[(size_t)b * H + n_glob], pe);
            atomicAdd(&sum_ce[(size_t)b * H + n_glob], pce);
        }
    }
}

// ---------------------------------------------------------------------------
// Kernel 3 (tiny, L2-resident weights; needs h_avg_sum from the GEMM pass):
//   fo[b][r] = sigmoid(g[b]·W_w[r] + (h_avg_sum[b]/W)·U_w[r] + U_b[r])
// ---------------------------------------------------------------------------
__global__ void k_fo(const float* __restrict__ g,
                     const float* __restrict__ h_avg_sum,
                     const float* __restrict__ W_w,
                     const float* __restrict__ U_w,
                     const float* __restrict__ U_b,
                     float* __restrict__ fo) {
    int tid = blockIdx.x * blockDim.x + threadIdx.x;
    if (tid >= B * H2) return;
    int b = tid / H2;
    int r = tid % H2;
    const float* grow = g + (size_t)b * H;
    const float* ha   = h_avg_sum + (size_t)b * H;
    const float* wr   = W_w + (size_t)r * H;
    const float* ur   = U_w + (size_t)r * H;
    constexpr float inv_w = 1.0f / (float)W;
    float acc = U_b[r];
#pragma unroll 4
    for (int k = 0; k < H; ++k)
        acc += grow[k] * wr[k] + (ha[k] * inv_w) * ur[k];
    fo[tid] = 1.0f / (1.0f + __expf(-acc));
}

// ---------------------------------------------------------------------------
// Kernel 4: new_c = f*c_g + sum_ce/sum_e ; new_g = o*tanh(new_c)
// d_out layout: [new_g (B*H) | new_c (B*H)]
// ---------------------------------------------------------------------------
__global__ void k_final(const float* __restrict__ fo,
                        const float* __restrict__ c_g,
                        const float* __restrict__ sum_e,
                        const float* __restrict__ sum_ce,
                        float* __restrict__ out) {
    int tid = blockIdx.x * blockDim.x + threadIdx.x;
    if (tid >= B * H) return;
    int b = tid / H, hc = tid % H;
    float f  = fo[(size_t)b * H2 + hc];
    float o  = fo[(size_t)b * H2 + H + hc];
    float nc = f * c_g[tid] + sum_ce[tid] / sum_e[tid];
    float ng = o * tanhf(nc);
    out[tid]         = ng;
    out[B * H + tid] = nc;
}

// ---------------------------------------------------------------------------
extern "C" void kernel_launch(void* const* d_in, const int* in_sizes, int n_in,
                              void* d_out, int out_size, void* d_ws, size_t ws_size,
                              hipStream_t stream) {
    const float* g   = (const float*)d_in[0];
    const float* c_g = (const float*)d_in[1];
    const float* h   = (const float*)d_in[2];
    const float* c   = (const float*)d_in[3];
    const float* W_w = (const float*)d_in[4];
    const float* w_w = (const float*)d_in[5];
    const float* U_w = (const float*)d_in[6];
    const float* U_b = (const float*)d_in[7];
    const float* u_w = (const float*)d_in[8];
    const float* u_b = (const float*)d_in[9];
    float* out = (float*)d_out;

    // Scratch layout (1.5 MB total, re-initialized every launch):
    float* ws        = (float*)d_ws;
    float* gw        = ws;                   // B*H
    float* h_avg_sum = gw + B * H;           // B*H
    float* sum_e     = h_avg_sum + B * H;    // B*H
    float* sum_ce    = sum_e + B * H;        // B*H
    float* fo        = sum_ce + B * H;       // B*2H

    k_gw_zero<<<(B * H) / 256, 256, 0, stream>>>(g, w_w, gw, h_avg_sum,
                                                 sum_e, sum_ce);
    dim3 grid(H / 64, W / 64, B);
    k_fw_wmma<<<grid, 256, 0, stream>>>(h, c, u_w, u_b, gw,
                                        h_avg_sum, sum_e, sum_ce);
    k_fo<<<(B * H2) / 256, 256, 0, stream>>>(g, h_avg_sum, W_w, U_w, U_b, fo);
    k_final<<<(B * H) / 256, 256, 0, stream>>>(fo, c_g, sum_e, sum_ce, out);
}